// MixtralSparseMoeBlock_37709812859143
// MI455X (gfx1250) — compile-verified
//
#include <hip/hip_runtime.h>

// MixtralSparseMoeBlock for MI455X (gfx1250, wave32).
// T=4096 tokens, H=2048 hidden, F=8192 ffn, E=8 experts, top-2 routing.
// Memory-bound on fp32 weights (~1.6GB @ 23.3TB/s); fp32 WMMA 16x16x4 keeps
// reference precision while matrix-op throughput is far above the roofline need.

#define T_TOK 4096
#define H_DIM 2048
#define F_DIM 8192
#define N_EXP 8

typedef __attribute__((ext_vector_type(2))) float v2f;
typedef __attribute__((ext_vector_type(8))) float v8f;

__device__ __forceinline__ v8f wmma_f32_k4(v2f a, v2f b, v8f c) {
  // D = A(16x4) x B(4x16) + C(16x16), fp32 in/out.
  // 8-arg form: (neg_a, A, neg_b, B, c_mod, C, reuse_a, reuse_b)
  return __builtin_amdgcn_wmma_f32_16x16x4_f32(false, a, false, b, (short)0, c,
                                               false, false);
}

// ---------------------------------------------------------------------------
// Router: logits -> softmax -> top2 -> renorm -> compacted per-expert lists.
// One wave per token; 8 waves (256 threads) per block.
// ---------------------------------------------------------------------------
__global__ __launch_bounds__(256) void moe_router(
    const float* __restrict__ X, const float* __restrict__ GW,
    int* __restrict__ counts, int* __restrict__ idxlist,
    float* __restrict__ wlist) {
  const int lane = threadIdx.x & 31;
  const int wave = threadIdx.x >> 5;
  const long t = (long)blockIdx.x * 8 + wave;

  float acc[N_EXP];
#pragma unroll
  for (int e = 0; e < N_EXP; ++e) acc[e] = 0.f;

  const float* xrow = X + t * H_DIM;
  for (int h = lane; h < H_DIM; h += 32) {
    const float x = xrow[h];
#pragma unroll
    for (int e = 0; e < N_EXP; ++e) acc[e] += x * GW[e * H_DIM + h];
  }
#pragma unroll
  for (int e = 0; e < N_EXP; ++e) {
#pragma unroll
    for (int off = 16; off > 0; off >>= 1)
      acc[e] += __shfl_xor(acc[e], off, 32);
  }

  if (lane == 0) {
    float m = acc[0];
#pragma unroll
    for (int e = 1; e < N_EXP; ++e) m = fmaxf(m, acc[e]);
    float p[N_EXP];
#pragma unroll
    for (int e = 0; e < N_EXP; ++e) p[e] = __expf(acc[e] - m);
    // softmax denominator cancels in the top-2 renormalization.
    int i0 = 0;
    for (int e = 1; e < N_EXP; ++e)
      if (p[e] > p[i0]) i0 = e;               // first index wins ties (jax top_k)
    int i1 = (i0 == 0) ? 1 : 0;
    for (int e = 0; e < N_EXP; ++e)
      if (e != i0 && p[e] > p[i1]) i1 = e;
    const float rs = 1.f / (p[i0] + p[i1]);
    const float w0 = p[i0] * rs, w1 = p[i1] * rs;

    int pos0 = atomicAdd(&counts[i0], 1);
    idxlist[i0 * T_TOK + pos0] = (int)t;
    wlist[i0 * T_TOK + pos0] = w0;
    int pos1 = atomicAdd(&counts[i1], 1);
    idxlist[i1 * T_TOK + pos1] = (int)t;
    wlist[i1 * T_TOK + pos1] = w1;
  }
}

// ---------------------------------------------------------------------------
// Up projection: G = silu(Xg @ W1^T) * (Xg @ W3^T) over gathered token rows.
// Block tile: 32 tokens x 256 F-cols; 8 waves, each wave 16x64 via 4 WMMA
// n-subtiles sharing the A fragment. K = H = 2048, staged 64-wide in LDS.
// ---------------------------------------------------------------------------
__global__ __launch_bounds__(256) void moe_up_gemm(
    const float* __restrict__ X, const float* __restrict__ W1,
    const float* __restrict__ W3, const int* __restrict__ count,
    const int* __restrict__ idxlist, float* __restrict__ G, int segStart) {
  const int cnt = *count;
  const int tBase = blockIdx.y * 32;           // row within segment
  if (segStart + tBase >= cnt) return;         // uniform early-out (pre-barrier)

  __shared__ float As[32 * 68];                // padded stride: no bank conflicts
  __shared__ int toks[32];

  const int tid = threadIdx.x;
  const int lane = tid & 31, wave = tid >> 5;
  const int wm = wave >> 2, wn = wave & 3;
  const int l15 = lane & 15, lhi = lane >> 4;
  const long nBase = (long)blockIdx.x * 256 + wn * 64;

  if (tid < 32) {
    const int lp = segStart + tBase + tid;
    toks[tid] = (lp < cnt) ? idxlist[lp] : -1;
  }
  __syncthreads();

  const v8f zero = {0.f, 0.f, 0.f, 0.f, 0.f, 0.f, 0.f, 0.f};
  v8f acc1[4] = {zero, zero, zero, zero};
  v8f acc3[4] = {zero, zero, zero, zero};

  for (int kk = 0; kk < H_DIM; kk += 64) {
#pragma unroll
    for (int i = 0; i < 8; ++i) {              // stage 32x64 A tile (gathered)
      const int idx = tid + i * 256;
      const int r = idx >> 6, c = idx & 63;
      const int tok = toks[r];
      As[r * 68 + c] = (tok >= 0) ? X[(long)tok * H_DIM + kk + c] : 0.f;
    }
    __syncthreads();

    if (kk + 64 < H_DIM) {                     // hint next weight slab into L2
      __builtin_prefetch(&W1[(nBase + l15) * (long)H_DIM + kk + 64], 0, 1);
      __builtin_prefetch(&W3[(nBase + l15) * (long)H_DIM + kk + 64], 0, 1);
    }

#pragma unroll 4
    for (int k4 = 0; k4 < 64; k4 += 4) {
      const v2f a = *(const v2f*)&As[(wm * 16 + l15) * 68 + k4 + 2 * lhi];
#pragma unroll
      for (int nt = 0; nt < 4; ++nt) {
        const long nrow = nBase + nt * 16 + l15;
        const long koff = (long)kk + k4 + 2 * lhi;
        const v2f b1 = *(const v2f*)&W1[nrow * H_DIM + koff];
        const v2f b3 = *(const v2f*)&W3[nrow * H_DIM + koff];
        acc1[nt] = wmma_f32_k4(a, b1, acc1[nt]);
        acc3[nt] = wmma_f32_k4(a, b3, acc3[nt]);
      }
    }
    __syncthreads();
  }

  // Fused SwiGLU epilogue: lane holds col (nBase+nt*16+l15), rows j + 8*lhi.
  const long gRowBase = tBase + wm * 16 + lhi * 8;
#pragma unroll
  for (int nt = 0; nt < 4; ++nt) {
    const long col = nBase + nt * 16 + l15;
#pragma unroll
    for (int j = 0; j < 8; ++j) {
      const float a1 = acc1[nt][j];
      const float g = (a1 / (1.f + __expf(-a1))) * acc3[nt][j];  // silu(a1)*a3
      G[(gRowBase + j) * (long)F_DIM + col] = g;
    }
  }
}

// ---------------------------------------------------------------------------
// Down projection + weighted scatter: out[token] += w * (G @ W2^T).
// Same tiling; K = F = 8192, N = H = 2048.
// ---------------------------------------------------------------------------
__global__ __launch_bounds__(256) void moe_down_gemm(
    const float* __restrict__ G, const float* __restrict__ W2,
    const int* __restrict__ count, const int* __restrict__ idxlist,
    const float* __restrict__ wlist, float* __restrict__ out, int segStart) {
  const int cnt = *count;
  const int tBase = blockIdx.y * 32;
  if (segStart + tBase >= cnt) return;

  __shared__ float As[32 * 68];

  const int tid = threadIdx.x;
  const int lane = tid & 31, wave = tid >> 5;
  const int wm = wave >> 2, wn = wave & 3;
  const int l15 = lane & 15, lhi = lane >> 4;
  const long nBase = (long)blockIdx.x * 256 + wn * 64;

  const v8f zero = {0.f, 0.f, 0.f, 0.f, 0.f, 0.f, 0.f, 0.f};
  v8f acc[4] = {zero, zero, zero, zero};

  for (int kk = 0; kk < F_DIM; kk += 64) {
#pragma unroll
    for (int i = 0; i < 8; ++i) {              // stage 32x64 tile of G
      const int idx = tid + i * 256;
      const int r = idx >> 6, c = idx & 63;
      As[r * 68 + c] = G[(long)(tBase + r) * F_DIM + kk + c];
    }
    __syncthreads();

    if (kk + 64 < F_DIM)
      __builtin_prefetch(&W2[(nBase + l15) * (long)F_DIM + kk + 64], 0, 1);

#pragma unroll 4
    for (int k4 = 0; k4 < 64; k4 += 4) {
      const v2f a = *(const v2f*)&As[(wm * 16 + l15) * 68 + k4 + 2 * lhi];
#pragma unroll
      for (int nt = 0; nt < 4; ++nt) {
        const long nrow = nBase + nt * 16 + l15;
        const v2f b = *(const v2f*)&W2[nrow * F_DIM + kk + k4 + 2 * lhi];
        acc[nt] = wmma_f32_k4(a, b, acc[nt]);
      }
    }
    __syncthreads();
  }

  // Scatter: each list position is a unique token within this expert, and
  // expert kernels serialize on the stream, so plain += is race-free.
#pragma unroll
  for (int nt = 0; nt < 4; ++nt) {
    const long col = nBase + nt * 16 + l15;
#pragma unroll
    for (int j = 0; j < 8; ++j) {
      const int lp = segStart + tBase + wm * 16 + lhi * 8 + j;
      if (lp < cnt) {
        const int tok = idxlist[lp];
        const float w = wlist[lp];
        out[(long)tok * H_DIM + col] += w * acc[nt][j];
      }
    }
  }
}

// ---------------------------------------------------------------------------
extern "C" void kernel_launch(void* const* d_in, const int* in_sizes, int n_in,
                              void* d_out, int out_size, void* d_ws,
                              size_t ws_size, hipStream_t stream) {
  const float* X  = (const float*)d_in[0];   // [T, H]
  const float* GW = (const float*)d_in[1];   // [E, H]
  const float* W1 = (const float*)d_in[2];   // [E, F, H]
  const float* W2 = (const float*)d_in[3];   // [E, H, F]
  const float* W3 = (const float*)d_in[4];   // [E, F, H]
  float* out = (float*)d_out;                // [T, H]

  // Workspace layout: [counts(64B)] [idxlist E*T i32] [wlist E*T f32] [G seg*F f32]
  char* ws = (char*)d_ws;
  int*   counts  = (int*)ws;
  int*   idxlist = (int*)(ws + 64);
  float* wlist   = (float*)(ws + 64 + (size_t)N_EXP * T_TOK * 4);
  float* G       = (float*)(ws + 64 + (size_t)2 * N_EXP * T_TOK * 4);
  const size_t headBytes = 64 + (size_t)2 * N_EXP * T_TOK * 4;

  long seg = 0;
  if (ws_size > headBytes)
    seg = (long)((ws_size - headBytes) / ((size_t)F_DIM * sizeof(float)));
  if (seg > T_TOK) seg = T_TOK;
  seg -= seg % 32;
  if (seg < 32) seg = 32;

  hipMemsetAsync(counts, 0, 64, stream);
  hipMemsetAsync(out, 0, (size_t)T_TOK * H_DIM * sizeof(float), stream);

  moe_router<<<T_TOK / 8, 256, 0, stream>>>(X, GW, counts, idxlist, wlist);

  const long FH = (long)F_DIM * H_DIM;
  for (int e = 0; e < N_EXP; ++e) {
    for (long s0 = 0; s0 < T_TOK; s0 += seg) {
      dim3 g1(F_DIM / 256, (unsigned)(seg / 32));
      moe_up_gemm<<<g1, 256, 0, stream>>>(
          X, W1 + (long)e * FH, W3 + (long)e * FH, counts + e,
          idxlist + (long)e * T_TOK, G, (int)s0);
      dim3 g2(H_DIM / 256, (unsigned)(seg / 32));
      moe_down_gemm<<<g2, 256, 0, stream>>>(
          G, W2 + (long)e * FH, counts + e, idxlist + (long)e * T_TOK,
          wlist + (long)e * T_TOK, out, (int)s0);
    }
  }
}